// BEMNA_V7_2_PhaseSpace_44117904064519
// MI455X (gfx1250) — compile-verified
//
#include <hip/hip_runtime.h>
#include <stdint.h>

// Problem geometry (matches reference GRID=(32,32,32), 6 channels).
// phi[c,x,y,z] linear: c*32768 + x*1024 + y*32 + z  (z fastest).
#define NCH    6
#define NCELL  32768
#define FRAME  (NCH * NCELL)          // 196608 floats per history frame
#define NWG    32                     // one 32x32 x-slab per workgroup
#define TPB    1024                   // 32*1024 = 32768 threads = 1 cell/thread

// ---------------------------------------------------------------------------
// Device-wide sense barrier (flags[0]=arrive count, flags[1]=generation).
// ---------------------------------------------------------------------------
__device__ __forceinline__ void gbar(unsigned* flags, unsigned round) {
  __threadfence();                    // release my global stores to agent scope
  __syncthreads();
  if (threadIdx.x == 0) {
    unsigned prev = __hip_atomic_fetch_add(&flags[0], 1u,
                        __ATOMIC_ACQ_REL, __HIP_MEMORY_SCOPE_AGENT);
    if (prev == NWG - 1) {
      __hip_atomic_store(&flags[0], 0u,
                        __ATOMIC_RELAXED, __HIP_MEMORY_SCOPE_AGENT);
      __hip_atomic_fetch_add(&flags[1], 1u,
                        __ATOMIC_ACQ_REL, __HIP_MEMORY_SCOPE_AGENT);
    } else {
      while (__hip_atomic_load(&flags[1], __ATOMIC_ACQUIRE,
                               __HIP_MEMORY_SCOPE_AGENT) < round) {
        __builtin_amdgcn_s_sleep(2);
      }
    }
  }
  __syncthreads();
  __threadfence();                    // acquire: refresh L0 before reading peers
}

__global__ void ca_init(unsigned* flags) {
  if (threadIdx.x == 0) { flags[0] = 0u; flags[1] = 0u; flags[2] = 0u; }
}

// ---------------------------------------------------------------------------
// Persistent kernel: up to T-1 barriered CA steps writing history in-place
// (history[t] IS phi_t), own-slab phi cached in LDS, async-LDS streaming tail.
// ---------------------------------------------------------------------------
__global__ __launch_bounds__(TPB, 1)
void ca_wave(const float* __restrict__ D,
             const int* __restrict__ sxp, const int* __restrict__ syp,
             const int* __restrict__ szp, const int* __restrict__ exp_,
             const int* __restrict__ eyp, const int* __restrict__ ezp,
             float* __restrict__ out, int T, unsigned* __restrict__ flags) {
  const int tid  = threadIdx.x;
  const int x0   = blockIdx.x;               // this WG's x-slab
  const int cell = (x0 << 10) + tid;         // tid = y*32 + z
  const int y = tid >> 5, z = tid & 31;

  const int st = ((*sxp) << 10) + ((*syp) << 5) + (*szp);
  const int et = ((*exp_) << 10) + ((*eyp) << 5) + (*ezp);

  __shared__ unsigned sdone;
  __shared__ __align__(16) float phiL[NCH * TPB];   // 24 KB: phi_t, own slab

  // ---- frame 0: impulse at (sx,sy,sz) on all 6 channels ----
  const float v0 = (cell == st) ? 1.0f : 0.0f;
#pragma unroll
  for (int o = 0; o < NCH; ++o) {
    out[(size_t)o * NCELL + cell] = v0;
    phiL[o * TPB + tid] = v0;
  }

  unsigned round = 0;
  gbar(flags, ++round);

  // Gather sources per output channel o (dest gets phi_out from src = dest-dir):
  //   o=0: x-1   o=1: x+1   o=2: y-1   o=3: y+1   o=4: z-1   o=5: z+1
  const int  xm = (x0 > 0)  ? x0 - 1 : 0;    // clamped (masked when invalid)
  const int  xp = (x0 < 31) ? x0 + 1 : 31;
  const bool vld[NCH]  = { x0 > 0, x0 < 31, y > 0, y < 31, z > 0, z < 31 };
  const int  lidx[NCH] = { tid, tid,
                           (y > 0)  ? tid - 32 : tid,
                           (y < 31) ? tid + 32 : tid,
                           (z > 0)  ? tid - 1  : tid,
                           (z < 31) ? tid + 1  : tid };
  const int  gxo[NCH]  = { xm, xp, x0, x0, x0, x0 };

  float last[NCH];
#pragma unroll
  for (int o = 0; o < NCH; ++o) last[o] = v0;
  int froz = -1;

  for (int t = 0; t + 1 < T; ++t) {
    if (tid == 0)
      sdone = __hip_atomic_load(&flags[2], __ATOMIC_ACQUIRE,
                                __HIP_MEMORY_SCOPE_AGENT);
    __syncthreads();
    if (sdone != 0u) { froz = t; break; }    // frames t+1..T-1 == frame t

    const float* __restrict__ cur = out + (size_t)t * FRAME;
    float*       __restrict__ nx  = out + (size_t)(t + 1) * FRAME;

    float ssum = 0.0f;
#pragma unroll
    for (int o = 0; o < NCH; ++o) {
      const int li = lidx[o];
      const int sb = (gxo[o] << 10) + li;    // src linear index (clamped)
      float acc = 0.0f;
      if (o < 2) {                            // cross-slab src: global reads
#pragma unroll
        for (int i = 0; i < NCH; ++i) {
          const float ph = cur[(size_t)i * NCELL + sb];
          const float dv = D[(size_t)(o * NCH + i) * NCELL + sb];
          acc = fmaf(dv + 0.95f, ph, acc);    // active_D = D + 0.95
        }
      } else {                                // in-slab src: LDS reads
#pragma unroll
        for (int i = 0; i < NCH; ++i) {
          const float ph = phiL[i * TPB + li];
          const float dv = D[(size_t)(o * NCH + i) * NCELL + sb];
          acc = fmaf(dv + 0.95f, ph, acc);
        }
      }
      float v = fminf(fmaxf(acc, 0.0f), 1.0f);  // clip(pn, 0, 1)
      v = vld[o] ? v : 0.0f;                    // zero-halo shift
      nx[(size_t)o * NCELL + cell] = v;         // history[t+1] written in place
      last[o] = v;
      ssum += v;
    }

    __syncthreads();                            // all phiL reads complete
#pragma unroll
    for (int o = 0; o < NCH; ++o) phiL[o * TPB + tid] = last[o]; // phi_{t+1}

    if (cell == et && ssum > 0.01f)
      __hip_atomic_store(&flags[2], 1u, __ATOMIC_RELEASE,
                         __HIP_MEMORY_SCOPE_AGENT);
    gbar(flags, ++round);
  }

  // ---- streaming tail: phiL already holds the frozen frame in channel-major
  //      chunk layout; replicate it to frames froz+1..T-1 with gfx1250 async
  //      LDS->global B128 stores (ASYNCcnt path), no barriers ----
  if (froz >= 0) {
    const unsigned long long ob = (unsigned long long)(uintptr_t)out;
    const unsigned ldsbase = (unsigned)(uintptr_t)&phiL[0]; // addr[31:0]=LDS off
    const unsigned slab4   = (unsigned)(x0 << 10) * 4u;
    for (int f = froz + 1; f < T; ++f) {
      const unsigned fb = (unsigned)((size_t)f * FRAME * 4u);
#pragma unroll
      for (int k = 0; k < 2; ++k) {
        const int u = tid + k * TPB;            // 16-byte unit id
        if (u < (NCH * TPB) / 4) {              // 1536 units per frame chunk
          const int ch = u >> 8;                // 256 units per channel
          const int lw = (u & 255) << 2;        // word within channel chunk
          const unsigned goff = fb + (unsigned)(ch * NCELL) * 4u + slab4
                              + (unsigned)lw * 4u;
          const unsigned loff = ldsbase + (unsigned)u * 16u;
          asm volatile("global_store_async_from_lds_b128 %0, %1, %2"
                       :: "v"(goff), "v"(loff), "s"(ob)
                       : "memory");
        }
      }
    }
    asm volatile("s_wait_asynccnt 0x0" ::: "memory");
  }
}

// ---------------------------------------------------------------------------
// Inputs (setup_inputs order): D f32[6,6,32,32,32], sx, sy, sz, ex, ey, ez,
// max_iterations (ints). Output: f32[T,6,32,32,32] flat.
// ---------------------------------------------------------------------------
extern "C" void kernel_launch(void* const* d_in, const int* in_sizes, int n_in,
                              void* d_out, int out_size, void* d_ws, size_t ws_size,
                              hipStream_t stream) {
  (void)in_sizes; (void)n_in; (void)ws_size;
  const float* D  = (const float*)d_in[0];
  const int*   sx = (const int*)d_in[1];
  const int*   sy = (const int*)d_in[2];
  const int*   sz = (const int*)d_in[3];
  const int*   ex = (const int*)d_in[4];
  const int*   ey = (const int*)d_in[5];
  const int*   ez = (const int*)d_in[6];
  float*    out   = (float*)d_out;
  unsigned* flags = (unsigned*)d_ws;

  const int T = out_size / FRAME;              // = max_iterations (300)
  if (T <= 0) return;

  ca_init<<<1, 32, 0, stream>>>(flags);
  ca_wave<<<NWG, TPB, 0, stream>>>(D, sx, sy, sz, ex, ey, ez, out, T, flags);
}